// Net_7705171329584
// MI455X (gfx1250) — compile-verified
//
#include <hip/hip_runtime.h>

#define N_NODES 100000
#define N_EDGES 1600000
#define DIM 64
#define N_GRAPHS 512

typedef __attribute__((ext_vector_type(2))) float v2f;
typedef __attribute__((ext_vector_type(8))) float v8f;

// ---------------------------------------------------------------------------
// Edge scatter: agg[dst] += h[src], one thread per (edge, 4-channel group).
// ---------------------------------------------------------------------------
__global__ __launch_bounds__(256) void scatter_add_kernel(
    const float* __restrict__ h, const int* __restrict__ edge_index,
    float* __restrict__ agg) {
  unsigned gid = blockIdx.x * 256u + threadIdx.x;
  if (gid >= (unsigned)N_EDGES * 16u) return;
  int e  = (int)(gid >> 4);
  int c4 = ((int)gid & 15) << 2;
  int src = edge_index[e];
  int dst = edge_index[N_EDGES + e];
  const float4 v = *(const float4*)(h + (size_t)src * DIM + c4);
  float* p = agg + (size_t)dst * DIM + c4;
  atomicAdd(p + 0, v.x);
  atomicAdd(p + 1, v.y);
  atomicAdd(p + 2, v.z);
  atomicAdd(p + 3, v.w);
}

// ---------------------------------------------------------------------------
// Fused GIN MLP: out = relu(relu((h+agg) @ w1 + b1) @ w2 + b2)
// One wave per 16-row tile; 8 waves (128 rows) per block.
// Uses V_WMMA_F32_16X16X4_F32 (exact fp32 path).
//
// A-frag layout (16x4 f32): lanes 0-15 hold {K=k0,k0+1} of row M=lane,
//                           lanes 16-31 hold {K=k0+2,k0+3} of row M=lane-16.
// B-frag layout (4x16 f32): lane l holds {w[ka][col], w[ka+1][col]},
//                           col = l&15, ka = k0 + 2*(l>>4).
// C/D layout (16x16 f32):   VGPR j: lanes 0-15 = (M=j, N=lane),
//                           lanes 16-31 = (M=j+8, N=lane-16).
// ---------------------------------------------------------------------------
__global__ __launch_bounds__(256) void gin_mlp_kernel(
    const float* __restrict__ h, const float* __restrict__ agg,
    const float* __restrict__ w1, const float* __restrict__ b1,
    const float* __restrict__ w2, const float* __restrict__ b2,
    float* __restrict__ out, int nrows) {
  __shared__ float lds_y[8 * 16 * DIM];  // 32 KB: per-wave 16x64 staging

  const int lane = threadIdx.x & 31;
  const int wave = threadIdx.x >> 5;
  const int half = lane >> 4;   // 0 | 1
  const int l16  = lane & 15;
  const int tile_base = blockIdx.x * 128 + wave * 16;
  int row  = tile_base + l16;
  int rrow = row < nrows ? row : (nrows - 1);  // clamp (keep EXEC all-ones)

  float* yt = lds_y + wave * (16 * DIM);

  // ---------------- GEMM 1: y = relu((h+agg) @ w1 + b1) ----------------
  v8f a0, a1, a2, a3;
  {
    float bb0 = b1[l16], bb1 = b1[16 + l16], bb2 = b1[32 + l16], bb3 = b1[48 + l16];
    #pragma unroll
    for (int j = 0; j < 8; ++j) { a0[j] = bb0; a1[j] = bb1; a2[j] = bb2; a3[j] = bb3; }
  }
  const float* zh = h   + (size_t)rrow * DIM;
  const float* za = agg + (size_t)rrow * DIM;
  #pragma unroll
  for (int k0 = 0; k0 < DIM; k0 += 4) {
    const int ka = k0 + 2 * half;
    float2 hv = *(const float2*)(zh + ka);
    float2 av = *(const float2*)(za + ka);
    v2f a; a.x = hv.x + av.x; a.y = hv.y + av.y;
    v2f b;
    b.x = w1[ka * DIM +  0 + l16]; b.y = w1[(ka + 1) * DIM +  0 + l16];
    a0 = __builtin_amdgcn_wmma_f32_16x16x4_f32(false, a, false, b, (short)0, a0, false, false);
    b.x = w1[ka * DIM + 16 + l16]; b.y = w1[(ka + 1) * DIM + 16 + l16];
    a1 = __builtin_amdgcn_wmma_f32_16x16x4_f32(false, a, false, b, (short)0, a1, false, false);
    b.x = w1[ka * DIM + 32 + l16]; b.y = w1[(ka + 1) * DIM + 32 + l16];
    a2 = __builtin_amdgcn_wmma_f32_16x16x4_f32(false, a, false, b, (short)0, a2, false, false);
    b.x = w1[ka * DIM + 48 + l16]; b.y = w1[(ka + 1) * DIM + 48 + l16];
    a3 = __builtin_amdgcn_wmma_f32_16x16x4_f32(false, a, false, b, (short)0, a3, false, false);
  }
  // ReLU + D-layout -> LDS (row major) for A-frag construction of GEMM 2.
  #pragma unroll
  for (int j = 0; j < 8; ++j) {
    int r = j + 8 * half;
    yt[r * DIM +  0 + l16] = fmaxf(a0[j], 0.0f);
    yt[r * DIM + 16 + l16] = fmaxf(a1[j], 0.0f);
    yt[r * DIM + 32 + l16] = fmaxf(a2[j], 0.0f);
    yt[r * DIM + 48 + l16] = fmaxf(a3[j], 0.0f);
  }
  __syncthreads();

  // ---------------- GEMM 2: out = relu(y @ w2 + b2) ----------------
  v8f c0, c1, c2, c3;
  {
    float bb0 = b2[l16], bb1 = b2[16 + l16], bb2 = b2[32 + l16], bb3 = b2[48 + l16];
    #pragma unroll
    for (int j = 0; j < 8; ++j) { c0[j] = bb0; c1[j] = bb1; c2[j] = bb2; c3[j] = bb3; }
  }
  const float* yr = yt + l16 * DIM;
  #pragma unroll
  for (int k0 = 0; k0 < DIM; k0 += 4) {
    const int ka = k0 + 2 * half;
    float2 yv = *(const float2*)(yr + ka);
    v2f a; a.x = yv.x; a.y = yv.y;
    v2f b;
    b.x = w2[ka * DIM +  0 + l16]; b.y = w2[(ka + 1) * DIM +  0 + l16];
    c0 = __builtin_amdgcn_wmma_f32_16x16x4_f32(false, a, false, b, (short)0, c0, false, false);
    b.x = w2[ka * DIM + 16 + l16]; b.y = w2[(ka + 1) * DIM + 16 + l16];
    c1 = __builtin_amdgcn_wmma_f32_16x16x4_f32(false, a, false, b, (short)0, c1, false, false);
    b.x = w2[ka * DIM + 32 + l16]; b.y = w2[(ka + 1) * DIM + 32 + l16];
    c2 = __builtin_amdgcn_wmma_f32_16x16x4_f32(false, a, false, b, (short)0, c2, false, false);
    b.x = w2[ka * DIM + 48 + l16]; b.y = w2[(ka + 1) * DIM + 48 + l16];
    c3 = __builtin_amdgcn_wmma_f32_16x16x4_f32(false, a, false, b, (short)0, c3, false, false);
  }
  // ReLU + predicated global store.
  #pragma unroll
  for (int j = 0; j < 8; ++j) {
    int gr = tile_base + j + 8 * half;
    if (gr < nrows) {
      float* o = out + (size_t)gr * DIM;
      o[ 0 + l16] = fmaxf(c0[j], 0.0f);
      o[16 + l16] = fmaxf(c1[j], 0.0f);
      o[32 + l16] = fmaxf(c2[j], 0.0f);
      o[48 + l16] = fmaxf(c3[j], 0.0f);
    }
  }
}

// ---------------------------------------------------------------------------
// global_add_pool: g[batch[n]] += h[n]
// ---------------------------------------------------------------------------
__global__ __launch_bounds__(256) void pool_kernel(
    const float* __restrict__ h, const int* __restrict__ batch,
    float* __restrict__ g) {
  unsigned gid = blockIdx.x * 256u + threadIdx.x;
  if (gid >= (unsigned)N_NODES * 16u) return;
  int n  = (int)(gid >> 4);
  int c4 = ((int)gid & 15) << 2;
  int bg = batch[n];
  const float4 v = *(const float4*)(h + (size_t)n * DIM + c4);
  float* p = g + (size_t)bg * DIM + c4;
  atomicAdd(p + 0, v.x);
  atomicAdd(p + 1, v.y);
  atomicAdd(p + 2, v.z);
  atomicAdd(p + 3, v.w);
}

// ---------------------------------------------------------------------------
// Head: out = relu(g @ lin1_w + lin1_b) @ lin2_w + lin2_b
// 4 blocks x 8 waves = 32 tiles of 16 graphs = 512.
// ---------------------------------------------------------------------------
__global__ __launch_bounds__(256) void head_kernel(
    const float* __restrict__ g,
    const float* __restrict__ w1, const float* __restrict__ b1,
    const float* __restrict__ w2, const float* __restrict__ b2,
    float* __restrict__ out) {
  __shared__ float lds_y[8 * 16 * DIM];

  const int lane = threadIdx.x & 31;
  const int wave = threadIdx.x >> 5;
  const int half = lane >> 4;
  const int l16  = lane & 15;
  const int tile_base = blockIdx.x * 128 + wave * 16;

  float* yt = lds_y + wave * (16 * DIM);

  v8f a0, a1, a2, a3;
  {
    float bb0 = b1[l16], bb1 = b1[16 + l16], bb2 = b1[32 + l16], bb3 = b1[48 + l16];
    #pragma unroll
    for (int j = 0; j < 8; ++j) { a0[j] = bb0; a1[j] = bb1; a2[j] = bb2; a3[j] = bb3; }
  }
  const float* gr = g + (size_t)(tile_base + l16) * DIM;
  #pragma unroll
  for (int k0 = 0; k0 < DIM; k0 += 4) {
    const int ka = k0 + 2 * half;
    float2 gv = *(const float2*)(gr + ka);
    v2f a; a.x = gv.x; a.y = gv.y;
    v2f b;
    b.x = w1[ka * DIM +  0 + l16]; b.y = w1[(ka + 1) * DIM +  0 + l16];
    a0 = __builtin_amdgcn_wmma_f32_16x16x4_f32(false, a, false, b, (short)0, a0, false, false);
    b.x = w1[ka * DIM + 16 + l16]; b.y = w1[(ka + 1) * DIM + 16 + l16];
    a1 = __builtin_amdgcn_wmma_f32_16x16x4_f32(false, a, false, b, (short)0, a1, false, false);
    b.x = w1[ka * DIM + 32 + l16]; b.y = w1[(ka + 1) * DIM + 32 + l16];
    a2 = __builtin_amdgcn_wmma_f32_16x16x4_f32(false, a, false, b, (short)0, a2, false, false);
    b.x = w1[ka * DIM + 48 + l16]; b.y = w1[(ka + 1) * DIM + 48 + l16];
    a3 = __builtin_amdgcn_wmma_f32_16x16x4_f32(false, a, false, b, (short)0, a3, false, false);
  }
  #pragma unroll
  for (int j = 0; j < 8; ++j) {
    int r = j + 8 * half;
    yt[r * DIM +  0 + l16] = fmaxf(a0[j], 0.0f);
    yt[r * DIM + 16 + l16] = fmaxf(a1[j], 0.0f);
    yt[r * DIM + 32 + l16] = fmaxf(a2[j], 0.0f);
    yt[r * DIM + 48 + l16] = fmaxf(a3[j], 0.0f);
  }
  __syncthreads();

  // lin2: 64 -> 1 dot product per graph (lanes 0-15 of each wave).
  if (half == 0) {
    float s = b2[0];
    const float* yr = yt + l16 * DIM;
    #pragma unroll
    for (int k = 0; k < DIM; ++k) s += yr[k] * w2[k];
    out[tile_base + l16] = s;
  }
}

// ---------------------------------------------------------------------------
extern "C" void kernel_launch(void* const* d_in, const int* in_sizes, int n_in,
                              void* d_out, int out_size, void* d_ws, size_t ws_size,
                              hipStream_t stream) {
  const float* x          = (const float*)d_in[0];
  const int*   edge_index = (const int*)d_in[1];
  const int*   batch      = (const int*)d_in[2];
  const float* cw1[3] = {(const float*)d_in[3],  (const float*)d_in[7],  (const float*)d_in[11]};
  const float* cb1[3] = {(const float*)d_in[4],  (const float*)d_in[8],  (const float*)d_in[12]};
  const float* cw2[3] = {(const float*)d_in[5],  (const float*)d_in[9],  (const float*)d_in[13]};
  const float* cb2[3] = {(const float*)d_in[6],  (const float*)d_in[10], (const float*)d_in[14]};
  const float* lin1_w = (const float*)d_in[15];
  const float* lin1_b = (const float*)d_in[16];
  const float* lin2_w = (const float*)d_in[17];
  const float* lin2_b = (const float*)d_in[18];

  float* ws  = (float*)d_ws;
  float* agg = ws;                                  // N_NODES*DIM
  float* hA  = agg + (size_t)N_NODES * DIM;         // N_NODES*DIM
  float* hB  = hA  + (size_t)N_NODES * DIM;         // N_NODES*DIM
  float* g   = hB  + (size_t)N_NODES * DIM;         // N_GRAPHS*DIM

  const unsigned scatter_blocks = ((unsigned)N_EDGES * 16u + 255u) / 256u;
  const unsigned mlp_blocks     = (N_NODES + 127) / 128;
  const unsigned pool_blocks    = ((unsigned)N_NODES * 16u + 255u) / 256u;

  const float* hin = x;
  float* pp[2] = {hA, hB};
  for (int layer = 0; layer < 3; ++layer) {
    hipMemsetAsync(agg, 0, (size_t)N_NODES * DIM * sizeof(float), stream);
    scatter_add_kernel<<<scatter_blocks, 256, 0, stream>>>(hin, edge_index, agg);
    float* hout = pp[layer & 1];
    gin_mlp_kernel<<<mlp_blocks, 256, 0, stream>>>(
        hin, agg, cw1[layer], cb1[layer], cw2[layer], cb2[layer], hout, N_NODES);
    hin = hout;
  }

  hipMemsetAsync(g, 0, (size_t)N_GRAPHS * DIM * sizeof(float), stream);
  pool_kernel<<<pool_blocks, 256, 0, stream>>>(hin, batch, g);
  head_kernel<<<4, 256, 0, stream>>>(g, lin1_w, lin1_b, lin2_w, lin2_b, (float*)d_out);
}